// SimpleGCN_47132971106898
// MI455X (gfx1250) — compile-verified
//
#include <hip/hip_runtime.h>

#define FD 64  // feature dim (in = hidden = out)

typedef __attribute__((ext_vector_type(2))) float v2f;
typedef __attribute__((ext_vector_type(8))) float v8f;

// ---------------------------------------------------------------- utilities
__global__ __launch_bounds__(256) void k_fill(float* __restrict__ p, float val, int n) {
    int i = blockIdx.x * 256 + threadIdx.x;
    if (i < n) p[i] = val;
}

__global__ __launch_bounds__(256) void k_count_deg(const long long* __restrict__ col,
                                                   float* __restrict__ deg, int E) {
    int e = blockIdx.x * 256 + threadIdx.x;
    if (e < E) unsafeAtomicAdd(&deg[(int)col[e]], 1.0f);  // global_atomic_add_f32, no return
}

__global__ __launch_bounds__(256) void k_rsqrt(float* __restrict__ p, int n) {
    int i = blockIdx.x * 256 + threadIdx.x;
    if (i < n) p[i] = rsqrtf(p[i]);  // deg >= 1 always (self-loop), so no zero guard needed
}

// ---------------------------------------------------------------- WMMA GEMM
// H[N,64] = X[N,64] @ W[64,64] using v_wmma_f32_16x16x4_f32.
// Block = 256 threads = 8 wave32; each wave computes one 16x16 output tile.
// Block tile: 32 rows x 64 cols (2 row-tiles x 4 col-tiles).
__global__ __launch_bounds__(256) void k_gemm64(const float* __restrict__ X,
                                                const float* __restrict__ W,
                                                float* __restrict__ H, int nrows) {
    const int wave  = threadIdx.x >> 5;   // 0..7 (uniform per wave)
    const int lane  = threadIdx.x & 31;
    const int rt    = wave >> 2;          // row tile in block: 0..1
    const int ct    = wave & 3;           // col tile in block: 0..3
    const int row_base = blockIdx.x * 32 + rt * 16;
    const int col_base = ct * 16;
    const int lhalf = lane >> 4;          // 0: lanes 0-15, 1: lanes 16-31
    const int lmod  = lane & 15;

    // Clamp A row with a select (keeps EXEC all-ones for WMMA); dup rows are
    // harmless since stores are guarded.
    int arow = row_base + lmod;
    arow = (arow < nrows) ? arow : (nrows - 1);
    const float* __restrict__ aptr = X + (size_t)arow * FD;

    v8f c = {};
#pragma unroll
    for (int kk = 0; kk < 16; ++kk) {
        const int k0 = kk * 4;
        // A 16x4 f32: lane (lmod) = row M; VGPR0/1 = K = 2*lhalf + {0,1}
        v2f a = *(const v2f*)(aptr + k0 + 2 * lhalf);           // aligned b64 load
        // B 4x16 f32: VGPR v, lane l -> K = k0 + 2*lhalf + v, N = lmod
        v2f b;
        b.x = W[(size_t)(k0 + 2 * lhalf + 0) * FD + col_base + lmod];
        b.y = W[(size_t)(k0 + 2 * lhalf + 1) * FD + col_base + lmod];
        c = __builtin_amdgcn_wmma_f32_16x16x4_f32(
                /*neg_a=*/false, a, /*neg_b=*/false, b,
                /*c_mod=*/(short)0, c, /*reuse_a=*/false, /*reuse_b=*/false);
    }

    // D 16x16 f32: VGPR r, lanes 0-15 -> M = r, lanes 16-31 -> M = r + 8; N = lmod
#pragma unroll
    for (int r = 0; r < 8; ++r) {
        int m = row_base + r + lhalf * 8;
        if (m < nrows) H[(size_t)m * FD + col_base + lmod] = c[r];
    }
}

// ---------------------------------------------------------------- edge scatter
// One wave32 per edge; each lane moves 2 consecutive floats (float2 gather,
// two f32 HW atomics to the L2-resident accumulation buffer).
__global__ __launch_bounds__(256) void k_aggregate(const float* __restrict__ H,
                                                   const long long* __restrict__ row,
                                                   const long long* __restrict__ col,
                                                   const float* __restrict__ dinv,
                                                   float* __restrict__ agg, int E) {
    int e = blockIdx.x * 8 + (threadIdx.x >> 5);
    if (e >= E) return;
    const int lane = threadIdx.x & 31;
    const int j = (int)row[e];   // source
    const int i = (int)col[e];   // target
    const float norm = dinv[i] * dinv[j];
    const float2 hv = *(const float2*)(H + (size_t)j * FD + lane * 2);
    float* dst = agg + (size_t)i * FD + lane * 2;
    unsafeAtomicAdd(dst + 0, norm * hv.x);
    unsafeAtomicAdd(dst + 1, norm * hv.y);
}

// ---------------------------------------------------------------- epilogue
// out = (relu?)(agg + dinv[i]^2 * h + bias). Safe when out aliases agg.
__global__ __launch_bounds__(256) void k_post(const float* __restrict__ agg,
                                              const float* __restrict__ h,
                                              const float* __restrict__ dinv,
                                              const float* __restrict__ bias,
                                              float* __restrict__ out, int n, int relu) {
    int i = blockIdx.x * 256 + threadIdx.x;
    if (i >= n) return;
    const int node = i >> 6;
    const int d    = i & 63;
    const float di = dinv[node];
    float v = agg[i] + di * di * h[i] + bias[d];
    if (relu) v = fmaxf(v, 0.0f);
    out[i] = v;
}

// ---------------------------------------------------------------- launcher
extern "C" void kernel_launch(void* const* d_in, const int* in_sizes, int n_in,
                              void* d_out, int out_size, void* d_ws, size_t ws_size,
                              hipStream_t stream) {
    const float*     x   = (const float*)d_in[0];
    const long long* ei  = (const long long*)d_in[1];  // int64 per reference
    const float*     W1  = (const float*)d_in[2];
    const float*     b1  = (const float*)d_in[3];
    const float*     W2  = (const float*)d_in[4];
    const float*     b2  = (const float*)d_in[5];
    float*           out = (float*)d_out;

    const int N = in_sizes[0] / FD;
    const int E = in_sizes[1] / 2;
    const int ND = N * FD;
    const long long* row = ei;       // sources j
    const long long* col = ei + E;   // targets i

    // workspace layout
    float* dinv = (float*)d_ws;              // N     (degrees -> dinv in place)
    float* h    = dinv + N;                  // N*FD  (GEMM output, reused for layer 2)
    float* agg  = h + (size_t)ND;            // N*FD  (layer-1 scatter accumulator)
    float* z1   = agg + (size_t)ND;          // N*FD  (layer-1 activations)

    const int thr = 256;
    const int bN  = (N  + thr - 1) / thr;
    const int bND = (ND + thr - 1) / thr;
    const int bE  = (E  + thr - 1) / thr;
    const int bEw = (E + 7) / 8;             // wave-per-edge blocks
    const int bG  = (N + 31) / 32;           // GEMM row-tile blocks

    // ---- normalization: deg = 1 (self loop) + indegree; dinv = rsqrt(deg)
    k_fill<<<bN,  thr, 0, stream>>>(dinv, 1.0f, N);
    k_fill<<<bND, thr, 0, stream>>>(agg, 0.0f, ND);
    k_fill<<<bND, thr, 0, stream>>>(out, 0.0f, ND);
    k_count_deg<<<bE, thr, 0, stream>>>(col, dinv, E);
    k_rsqrt<<<bN, thr, 0, stream>>>(dinv, N);

    // ---- layer 1
    k_gemm64<<<bG, thr, 0, stream>>>(x, W1, h, N);
    k_aggregate<<<bEw, thr, 0, stream>>>(h, row, col, dinv, agg, E);
    k_post<<<bND, thr, 0, stream>>>(agg, h, dinv, b1, z1, ND, /*relu=*/1);

    // ---- layer 2 (aggregate directly into d_out, epilogue in place)
    k_gemm64<<<bG, thr, 0, stream>>>(z1, W2, h, N);
    k_aggregate<<<bEw, thr, 0, stream>>>(h, row, col, dinv, out, E);
    k_post<<<bND, thr, 0, stream>>>(out, h, dinv, b2, out, ND, /*relu=*/0);
}